// Attn_30520037606032
// MI455X (gfx1250) — compile-verified
//
#include <hip/hip_runtime.h>

typedef __attribute__((ext_vector_type(2))) float v2f;
typedef __attribute__((ext_vector_type(4))) float v4f;
typedef __attribute__((ext_vector_type(8))) float v8f;

namespace {
constexpr int N = 32, M = 100, L = 8192, D = 128;
constexpr int MP    = 112;       // M padded to multiple of 16
constexpr int ROWP  = D + 4;     // 132-float LDS row pitch -> bank-conflict-free B reads
constexpr int NCH   = D / 4;     // 32 K=4 chunks per 16x16 WMMA tile
constexpr int MTILES = MP / 16;  // 7
constexpr int WAVES = 8;         // waves per block -> 128 l rows per block
}

__global__ __launch_bounds__(256)
void attn_loc_kernel(const float* __restrict__ self_attn,
                     const float* __restrict__ self_delta,
                     const float* __restrict__ emb_table,
                     const float* __restrict__ value_w,
                     float* __restrict__ out) {
  __shared__ float lds_attn[MP * ROWP];
  __shared__ float lds_w[MP];

  const int t  = threadIdx.x;
  const int n  = blockIdx.x >> 6;   // 64 blocks per batch element
  const int lg = blockIdx.x & 63;

  // ---- stage self_attn[n] (100x128 f32) into LDS, rows padded/zeroed to 112 ----
  const v4f* an = (const v4f*)(self_attn + (size_t)n * M * D);
  for (int idx = t; idx < (M * D) / 4; idx += 256) {
    int row = idx >> 5, c4 = idx & 31;
    *(v4f*)&lds_attn[row * ROWP + c4 * 4] = an[idx];
  }
  const v4f z4 = {};
  for (int idx = t; idx < ((MP - M) * D) / 4; idx += 256) {
    int row = M + (idx >> 5), c4 = idx & 31;
    *(v4f*)&lds_attn[row * ROWP + c4 * 4] = z4;
  }
  if (t < MP) lds_w[t] = (t < M) ? value_w[t] : 0.f;
  __syncthreads();

  const int wave = t >> 5, lane = t & 31;
  const int lr = lane & 15, hi = lane >> 4;
  const int l_base = (lg * WAVES + wave) * 16;

  // ---- A tile: emb_cand rows l_base..+15; f32 A layout: lane=row, K=4c+2hi,+1 ----
  const float* emb = emb_table + D;  // emb_cand = emb_table[1:]
  const v2f* ap = (const v2f*)(emb + (size_t)(l_base + lr) * D + hi * 2);
  v2f a[NCH];
#pragma unroll
  for (int c = 0; c < NCH; ++c) a[c] = ap[2 * c];

  float acc[8];
#pragma unroll
  for (int r = 0; r < 8; ++r) acc[r] = 0.f;

  for (int mt = 0; mt < MTILES; ++mt) {
    const int m = mt * 16 + lr;

    // prefetch next m-tile's delta slab while we compute this one
    if (mt + 1 < MTILES) {
      const int m2 = m + 16;
      if (m2 < M)
        __builtin_prefetch(self_delta + (((size_t)n * M + m2) * L + l_base + hi * 8) * 2, 0, 1);
    }

    // ---- B tile from LDS: lane=col(m), K=4c+2hi,+1 (mirrors A layout) ----
    const v2f* bp = (const v2f*)&lds_attn[m * ROWP + hi * 2];
    v2f b[NCH];
#pragma unroll
    for (int c = 0; c < NCH; ++c) b[c] = bp[2 * c];

    // ---- S[l,m] = sum_d emb[l,d] * attn[m,d] : 32 chained f32 WMMAs ----
    v8f ct = {};
#pragma unroll
    for (int c = 0; c < NCH; ++c)
      ct = __builtin_amdgcn_wmma_f32_16x16x4_f32(false, a[c], false, b[c],
                                                 (short)0, ct, false, false);

    // ---- delta pairs: streamed once -> nontemporal; contiguous 64B per lane ----
    v2f dp[8];
    if (m < M) {
      const v2f* dptr =
          (const v2f*)(self_delta + (((size_t)n * M + m) * L + l_base + hi * 8) * 2);
#pragma unroll
      for (int r = 0; r < 8; ++r) dp[r] = __builtin_nontemporal_load(dptr + r);
    } else {
#pragma unroll
      for (int r = 0; r < 8; ++r) { dp[r].x = 0.f; dp[r].y = 0.f; }
    }

    const float w = lds_w[m];
    // C layout: VGPR r holds l = l_base + r + 8*hi, lane holds column m
#pragma unroll
    for (int r = 0; r < 8; ++r) acc[r] += ct[r] * (dp[r].x + dp[r].y) * w;
  }

  // ---- reduce the 16 m-lanes per half-wave; lanes {0,16} hold final rows ----
#pragma unroll
  for (int r = 0; r < 8; ++r) {
    float v = acc[r];
    v += __shfl_xor(v, 8, 16);
    v += __shfl_xor(v, 4, 16);
    v += __shfl_xor(v, 2, 16);
    v += __shfl_xor(v, 1, 16);
    acc[r] = v;
  }
  if (lr == 0) {
    float* o = out + (size_t)n * L + l_base + hi * 8;
#pragma unroll
    for (int r = 0; r < 8; ++r) o[r] = acc[r];
  }
}

extern "C" void kernel_launch(void* const* d_in, const int* in_sizes, int n_in,
                              void* d_out, int out_size, void* d_ws, size_t ws_size,
                              hipStream_t stream) {
  const float* self_attn  = (const float*)d_in[0];
  const float* self_delta = (const float*)d_in[1];
  const float* emb_table  = (const float*)d_in[2];
  const float* value_w    = (const float*)d_in[3];
  (void)in_sizes; (void)n_in; (void)d_ws; (void)ws_size; (void)out_size;
  float* out = (float*)d_out;

  dim3 grid(N * (L / (16 * WAVES)));  // 32 * 64 = 2048 blocks, 8 waves each
  attn_loc_kernel<<<grid, 256, 0, stream>>>(self_attn, self_delta, emb_table,
                                            value_w, out);
}